// weight_generator_36876589203657
// MI455X (gfx1250) — compile-verified
//
#include <hip/hip_runtime.h>
#include <hip/hip_bf16.h>
#include <stdint.h>

#define BDIM 256
#define KNN  12

// gfx1250 async global->LDS path (ASYNCcnt). Guarded: fall back to direct
// loads if the builtin is not declared on this toolchain.
#if defined(__HIP_DEVICE_COMPILE__) && defined(__gfx1250__) && \
    __has_builtin(__builtin_amdgcn_global_load_async_to_lds_b128)
#define USE_ASYNC_LDS 1
#else
#define USE_ASYNC_LDS 0
#endif

#if USE_ASYNC_LDS
typedef int v4i __attribute__((ext_vector_type(4)));

__device__ __forceinline__ void async_b128_to_lds(const void* gsrc, void* lds_dst) {
  // (v4i AS1* gsrc, v4i AS3* lds, imm offset, imm cpol)
  __builtin_amdgcn_global_load_async_to_lds_b128(
      (__attribute__((address_space(1))) v4i*)(uintptr_t)gsrc,
      (__attribute__((address_space(3))) v4i*)lds_dst,
      0, 0);
}
__device__ __forceinline__ void wait_async0() {
#if __has_builtin(__builtin_amdgcn_s_wait_asynccnt)
  __builtin_amdgcn_s_wait_asynccnt(0);
#else
  asm volatile("s_wait_asynccnt 0" ::: "memory");
#endif
}
#endif

__global__ __launch_bounds__(BDIM) void knn_softmax_weight_kernel(
    const float* __restrict__ xyz1,    // [B, N, 3]
    const float* __restrict__ query,   // [B*M, 3]
    const int*   __restrict__ pidx,    // [B*M, 12]
    float*       __restrict__ out,     // [B*M, 12]
    int Mrows,                         // M
    int Npts,                          // N
    int Rtot)                          // B*M
{
  const int t    = threadIdx.x;
  const int row0 = blockIdx.x * BDIM;
  const int row  = row0 + t;

#if USE_ASYNC_LDS
  __shared__ int   s_idx[BDIM * KNN]; // 12 KB: linear mirror of the global tile
  __shared__ float s_q[BDIM * 3];     //  3 KB

  // --- stage index tile: BDIM*KNN*4 = 12288 B = 768 x 16B chunks (3/thread) ---
  {
    const char*     gbase      = (const char*)pidx;
    const long long totalBytes = (long long)Rtot * (KNN * 4);   // multiple of 16
    const long long tileBase   = (long long)row0 * (KNN * 4);
    char*           lbase      = (char*)s_idx;
#pragma unroll
    for (int c = 0; c < 3; ++c) {
      long long off = tileBase + (long long)c * (BDIM * 16) + (long long)t * 16;
      if (off + 16 > totalBytes) off = totalBytes - 16;          // tail clamp
      async_b128_to_lds(gbase + off, lbase + c * (BDIM * 16) + t * 16);
    }
  }
  // --- stage query tile: BDIM*12 = 3072 B = 192 x 16B chunks ---
  {
    const char*     gbase      = (const char*)query;
    const long long totalBytes = (long long)Rtot * 12;           // multiple of 16
    const long long tileBase   = (long long)row0 * 12;
    if (t < (BDIM * 12) / 16) {                                  // t < 192
      long long off = tileBase + (long long)t * 16;
      if (off + 16 > totalBytes) off = totalBytes - 16;
      async_b128_to_lds(gbase + off, (char*)s_q + t * 16);
    }
  }
  wait_async0();
  __syncthreads();
#endif

  if (row < Rtot) {
    float qx, qy, qz;
#if USE_ASYNC_LDS
    qx = s_q[t * 3 + 0];
    qy = s_q[t * 3 + 1];
    qz = s_q[t * 3 + 2];
#else
    qx = query[(size_t)row * 3 + 0];
    qy = query[(size_t)row * 3 + 1];
    qz = query[(size_t)row * 3 + 2];
#endif
    const int    b  = row / Mrows;                 // batch index; runtime divisor (kernel arg)
    const float* xb = xyz1 + (size_t)b * Npts * 3;

    float s[KNN];
    float mx = -1e30f;
#pragma unroll
    for (int k = 0; k < KNN; ++k) {
#if USE_ASYNC_LDS
      const int id = s_idx[t * KNN + k];
#else
      const int id = pidx[(size_t)row * KNN + k];
#endif
      const float* p  = xb + (size_t)id * 3;       // random gather, L2-resident
      const float  dx = p[0] - qx;
      const float  dy = p[1] - qy;
      const float  dz = p[2] - qz;
      const float  d  = sqrtf(dx * dx + dy * dy + dz * dz);
      const float  v  = -200.0f * d;
      s[k] = v;
      mx   = fmaxf(mx, v);
    }

    float e[KNN];
    float sum = 0.0f;
#pragma unroll
    for (int k = 0; k < KNN; ++k) {
      const float ek = __expf(s[k] - mx);          // v_exp_f32
      e[k] = ek;
      sum += ek;
    }
    const float inv = 0.5f / sum;

    float w[KNN];
#pragma unroll
    for (int k = 0; k < KNN; ++k) w[k] = e[k] * inv;
    w[0] += 0.5f;

    // 48 B per row, 16 B aligned -> three b128 stores
    float4* o = (float4*)(out + (size_t)row * KNN);
    o[0] = make_float4(w[0], w[1], w[2],  w[3]);
    o[1] = make_float4(w[4], w[5], w[6],  w[7]);
    o[2] = make_float4(w[8], w[9], w[10], w[11]);
  }
}

extern "C" void kernel_launch(void* const* d_in, const int* in_sizes, int n_in,
                              void* d_out, int out_size, void* d_ws, size_t ws_size,
                              hipStream_t stream) {
  (void)n_in; (void)out_size; (void)d_ws; (void)ws_size;
  // setup_inputs order:
  // 0: feature1 (dead), 1: xyz1, 2: query_xyz, 3: error (dead),
  // 4: point_index, 5: W_prefix (dead), 6: b_prefix (dead)
  const float* xyz1  = (const float*)d_in[1];
  const float* query = (const float*)d_in[2];
  const int*   pidx  = (const int*)d_in[4];
  float*       out   = (float*)d_out;

  const int Bc    = 4;                    // batch dim from reference
  const int Rtot  = in_sizes[2] / 3;      // B*M
  const int Npts  = in_sizes[1] / (3 * Bc);
  const int Mrows = Rtot / Bc;

  const int blocks = (Rtot + BDIM - 1) / BDIM;
  hipLaunchKernelGGL(knn_softmax_weight_kernel, dim3(blocks), dim3(BDIM), 0, stream,
                     xyz1, query, pidx, out, Mrows, Npts, Rtot);
}